// Pinn_62603443306877
// MI455X (gfx1250) — compile-verified
//
#include <hip/hip_runtime.h>

// ---------------- problem constants ----------------
#define DIM    128
#define INTER  512
#define NBLK   8
#define BATCH  4096
#define NJ     24                 // jet components per sample
#define SPG    4                  // samples per workgroup
#define NCOL   (NJ * SPG)         // 96 columns = 6 tiles of 16
#define NT     (NCOL / 16)        // 6 N-tiles
#define HSTRIDE 132               // H panel column stride (floats): 4 mod 64 banks, 8B aligned
#define ASTRIDE 516               // A panel column stride (floats): 512 rows + pad
#define LDS_FLOATS (NCOL * HSTRIDE + NCOL * ASTRIDE)   // ~243 KB

typedef __attribute__((ext_vector_type(2))) float v2f;
typedef __attribute__((ext_vector_type(8))) float v8f;

__device__ __forceinline__ v8f wmma4(v2f a, v2f b, v8f c) {
  // D = A(16x4) * B(4x16) + C, fp32 WMMA
  return __builtin_amdgcn_wmma_f32_16x16x4_f32(false, a, false, b, (short)0, c,
                                               false, false);
}

// symmetric pair index for i<=j: i*(7-i)/2 + j
__device__ __forceinline__ int symidx(int i, int j) {
  int a = i < j ? i : j;
  int b = i < j ? j : i;
  return a * (7 - a) / 2 + b;
}

__global__ void __launch_bounds__(256, 1)
pinn_jet_kernel(const float* __restrict__ x, const float* __restrict__ y,
                const float* __restrict__ z, const float* __restrict__ t,
                const float* __restrict__ u_in, const float* __restrict__ v_in,
                const float* __restrict__ w_in,
                const float* __restrict__ Wf, const float* __restrict__ bf,
                const float* __restrict__ W1, const float* __restrict__ b1,
                const float* __restrict__ W2, const float* __restrict__ b2,
                const float* __restrict__ Wl, const float* __restrict__ bl,
                const float* __restrict__ lam1p, const float* __restrict__ lam2p,
                float* __restrict__ out)
{
  extern __shared__ float lds[];
  float* Hbuf = lds;                     // DIM   x NCOL jet panel    [col][row]
  float* Abuf = lds + NCOL * HSTRIDE;    // INTER x NCOL z/act panel  [col][row]

  const int tid  = threadIdx.x;
  const int lane = tid & 31;
  const int wv   = tid >> 5;             // 8 waves
  const int mg   = wv & 3;               // M-group (covers 2 M-tiles)
  const int ng   = wv >> 2;              // N-group (covers 3 N-tiles)
  const int b0   = blockIdx.x * SPG;

  const int mrow  = lane & 15;           // A: row-in-tile, B/C: col-in-tile
  const int khalf = lane >> 4;           // 0 | 1
  const int rbase = khalf << 3;          // C-layout M base (0 or 8)
  const int koff  = khalf << 1;          // A/B K offset (0 or 2)

  // ---------------- initial jet: h = Wf s + bf, s = x/5 - 1 ----------------
  for (int q = 0; q < 2; ++q) {
    int item = tid + q * 256;            // 512 = 128 rows * 4 samples
    int row  = item & (DIM - 1);
    int s    = item >> 7;
    int b    = b0 + s;
    float sc0 = x[b] * 0.2f - 1.0f;
    float sc1 = y[b] * 0.2f - 1.0f;
    float sc2 = z[b] * 0.2f - 1.0f;
    float sc3 = t[b] * 0.2f - 1.0f;
    const float* wr = Wf + row * 4;
    float h0 = wr[0]*sc0 + wr[1]*sc1 + wr[2]*sc2 + wr[3]*sc3 + bf[row];
    float* hc = Hbuf + (s * NJ) * HSTRIDE + row;
    hc[0] = h0;
    #pragma unroll
    for (int i = 0; i < 4; ++i) hc[(1 + i) * HSTRIDE] = 0.2f * wr[i];
    #pragma unroll
    for (int j = 5; j < NJ; ++j) hc[j * HSTRIDE] = 0.0f;
  }
  __syncthreads();

  // ---------------- residual blocks ----------------
  for (int bk = 0; bk < NBLK; ++bk) {
    const float* W1b = W1 + bk * INTER * DIM;
    const float* W2b = W2 + bk * DIM * INTER;
    const float* b1b = b1 + bk * INTER;
    const float* b2b = b2 + bk * DIM;

    // ===== phase 1: A_full(512 x NCOL) = tanh-jet( W1 @ H + b1 ) into Abuf =====
    for (int ch = 0; ch < INTER / DIM; ++ch) {
      v8f zacc[2][3];
      #pragma unroll
      for (int mi = 0; mi < 2; ++mi)
        #pragma unroll
        for (int ni = 0; ni < 3; ++ni)
          #pragma unroll
          for (int r = 0; r < 8; ++r) zacc[mi][ni][r] = 0.0f;

      const float* w1r0 = W1b + (ch * DIM + (mg * 2 + 0) * 16 + mrow) * DIM;
      const float* w1r1 = W1b + (ch * DIM + (mg * 2 + 1) * 16 + mrow) * DIM;
      #pragma unroll 2
      for (int kk = 0; kk < DIM / 4; ++kk) {
        int k0 = kk * 4 + koff;
        v2f a0 = *(const v2f*)(w1r0 + k0);
        v2f a1 = *(const v2f*)(w1r1 + k0);
        #pragma unroll
        for (int ni = 0; ni < 3; ++ni) {
          v2f bb = *(const v2f*)(Hbuf + ((ng * 3 + ni) * 16 + mrow) * HSTRIDE + k0);
          zacc[0][ni] = wmma4(a0, bb, zacc[0][ni]);
          zacc[1][ni] = wmma4(a1, bb, zacc[1][ni]);
        }
      }
      // spill z tiles into Abuf (paired rows -> v2f stores)
      #pragma unroll
      for (int mi = 0; mi < 2; ++mi) {
        int rowb = ch * DIM + (mg * 2 + mi) * 16 + rbase;
        #pragma unroll
        for (int ni = 0; ni < 3; ++ni) {
          int col = (ng * 3 + ni) * 16 + mrow;
          #pragma unroll
          for (int r = 0; r < 8; r += 2) {
            v2f vv = { zacc[mi][ni][r], zacc[mi][ni][r + 1] };
            *(v2f*)(Abuf + col * ASTRIDE + rowb + r) = vv;
          }
        }
      }
      __syncthreads();

      // ---- tanh jet nonlinearity for this chunk, per (neuron,sample), in place ----
      for (int q = 0; q < 2; ++q) {
        int item = tid + q * 256;        // 512 = 128 rows * 4 samples
        int row  = item & (DIM - 1);
        int s    = item >> 7;
        float* zc = Abuf + (s * NJ) * ASTRIDE + ch * DIM + row;
        float zv = zc[0] + b1b[ch * DIM + row];
        float g[4], H[10], T[3][3];
        #pragma unroll
        for (int i = 0; i < 4; ++i)  g[i] = zc[(1 + i) * ASTRIDE];
        #pragma unroll
        for (int p = 0; p < 10; ++p) H[p] = zc[(5 + p) * ASTRIDE];
        #pragma unroll
        for (int i = 0; i < 3; ++i)
          #pragma unroll
          for (int j = 0; j < 3; ++j) T[i][j] = zc[(15 + i * 3 + j) * ASTRIDE];

        float a  = tanhf(zv);
        float t1 = 1.0f - a * a;                    // tanh'
        float t2 = -2.0f * a * t1;                  // tanh''
        float t3 = t1 * (4.0f * a * a - 2.0f * t1); // tanh'''

        zc[0] = a;
        #pragma unroll
        for (int i = 0; i < 4; ++i) zc[(1 + i) * ASTRIDE] = t1 * g[i];
        {
          int p = 0;
          #pragma unroll
          for (int i = 0; i < 4; ++i)
            #pragma unroll
            for (int j = i; j < 4; ++j, ++p)
              zc[(5 + p) * ASTRIDE] = t2 * g[i] * g[j] + t1 * H[p];
        }
        #pragma unroll
        for (int i = 0; i < 3; ++i)
          #pragma unroll
          for (int j = 0; j < 3; ++j) {
            float gj = g[j];
            zc[(15 + i * 3 + j) * ASTRIDE] =
                t3 * g[i] * gj * gj +
                t2 * (H[symidx(j, j)] * g[i] + 2.0f * H[symidx(i, j)] * gj) +
                t1 * T[i][j];
          }
      }
      __syncthreads();
    }

    // ===== phase 2: Hnew = H + W2 @ A + b2 (K = 512) =====
    v8f hacc[2][3];
    #pragma unroll
    for (int mi = 0; mi < 2; ++mi)
      #pragma unroll
      for (int ni = 0; ni < 3; ++ni)
        #pragma unroll
        for (int r = 0; r < 8; ++r)
          hacc[mi][ni][r] = Hbuf[((ng * 3 + ni) * 16 + mrow) * HSTRIDE +
                                 (mg * 2 + mi) * 16 + rbase + r];

    const float* w2r0 = W2b + ((mg * 2 + 0) * 16 + mrow) * INTER;
    const float* w2r1 = W2b + ((mg * 2 + 1) * 16 + mrow) * INTER;
    #pragma unroll 2
    for (int kk = 0; kk < INTER / 4; ++kk) {
      int k0 = kk * 4 + koff;
      v2f a0 = *(const v2f*)(w2r0 + k0);
      v2f a1 = *(const v2f*)(w2r1 + k0);
      #pragma unroll
      for (int ni = 0; ni < 3; ++ni) {
        v2f bb = *(const v2f*)(Abuf + ((ng * 3 + ni) * 16 + mrow) * ASTRIDE + k0);
        hacc[0][ni] = wmma4(a0, bb, hacc[0][ni]);
        hacc[1][ni] = wmma4(a1, bb, hacc[1][ni]);
      }
    }
    __syncthreads();   // all waves finished reading Hbuf (init) and Abuf

    // write updated jet panel (add b2 to value component only)
    #pragma unroll
    for (int mi = 0; mi < 2; ++mi) {
      #pragma unroll
      for (int ni = 0; ni < 3; ++ni) {
        int col = (ng * 3 + ni) * 16 + mrow;
        float bsel = (col % NJ == 0) ? 1.0f : 0.0f;
        #pragma unroll
        for (int r = 0; r < 8; r += 2) {
          int row = (mg * 2 + mi) * 16 + rbase + r;
          v2f vv = { hacc[mi][ni][r]     + bsel * b2b[row],
                     hacc[mi][ni][r + 1] + bsel * b2b[row + 1] };
          *(v2f*)(Hbuf + col * HSTRIDE + row) = vv;
        }
      }
    }
    __syncthreads();
  }

  // ---------------- final layer: out_jet[o][col] = Wl[o,:] . H[:,col] ----------------
  for (int job = tid; job < 3 * NCOL; job += 256) {
    int o   = job / NCOL;
    int col = job - o * NCOL;
    const float* wl = Wl + o * DIM;
    float acc = 0.0f;
    for (int r = 0; r < DIM; ++r) acc += wl[r] * Hbuf[col * HSTRIDE + r];
    if (col % NJ == 0) acc += bl[o];
    Abuf[o * NCOL + col] = acc;          // reuse A panel as out_jet[3][NCOL]
  }
  __syncthreads();

  // ---------------- residuals, preds, loss (1 thread / sample) ----------------
  if (tid < SPG) {
    const int s = tid, b = b0 + s, base = s * NJ;
    const float* OJ = Abuf;
    const float lam1 = lam1p[0], lam2 = lam2p[0];

    float Pg[4], pg[4], wg4[4];
    #pragma unroll
    for (int i = 0; i < 4; ++i) {
      Pg[i]  = OJ[base + 1 + i];
      pg[i]  = OJ[NCOL + base + 1 + i];
      wg4[i] = OJ[2 * NCOL + base + 1 + i];
    }
    float p_pred = OJ[NCOL + base];
    float w_pred = OJ[2 * NCOL + base];
    float u_pred = Pg[1] - Pg[2];
    float v_pred = -Pg[0] + Pg[2];

    float Ju[4], Jv[4];
    #pragma unroll
    for (int k = 0; k < 4; ++k) {
      Ju[k] = OJ[base + 5 + symidx(1, k)] - OJ[base + 5 + symidx(2, k)];
      Jv[k] = -OJ[base + 5 + symidx(0, k)] + OJ[base + 5 + symidx(2, k)];
    }
    float u_lap = 0.0f, v_lap = 0.0f, w_lap = 0.0f;
    #pragma unroll
    for (int j = 0; j < 3; ++j) {
      u_lap += OJ[base + 15 + 1 * 3 + j] - OJ[base + 15 + 2 * 3 + j];
      v_lap += -OJ[base + 15 + 0 * 3 + j] + OJ[base + 15 + 2 * 3 + j];
      w_lap += OJ[2 * NCOL + base + 5 + symidx(j, j)];
    }
    float f_u = lam1 * (Ju[3] + u_pred * Ju[0] + v_pred * Ju[1] + w_pred * Ju[2])
                + pg[0] - lam2 * u_lap;
    float f_v = lam1 * (Jv[3] + u_pred * Jv[0] + v_pred * Jv[1] + w_pred * Jv[2])
                + pg[1] - lam2 * v_lap;
    float f_w = lam1 * (wg4[3] + u_pred * wg4[0] + v_pred * wg4[1] + w_pred * wg4[2])
                + pg[2] - lam2 * w_lap;

    out[b * 4 + 0] = p_pred;
    out[b * 4 + 1] = u_pred;
    out[b * 4 + 2] = v_pred;
    out[b * 4 + 3] = w_pred;

    float du = u_pred - u_in[b], dv = v_pred - v_in[b], dw = w_pred - w_in[b];
    float lc = (du * du + dv * dv + dw * dw + f_u * f_u + f_v * f_v + f_w * f_w)
               * (1.0f / (float)BATCH);
    atomicAdd(out + 4 * BATCH, lc);
  }
}

__global__ void zero_loss_kernel(float* p) {
  if (threadIdx.x == 0 && blockIdx.x == 0) p[0] = 0.0f;
}

extern "C" void kernel_launch(void* const* d_in, const int* in_sizes, int n_in,
                              void* d_out, int out_size, void* d_ws, size_t ws_size,
                              hipStream_t stream) {
  const float* x    = (const float*)d_in[0];
  const float* y    = (const float*)d_in[1];
  const float* z    = (const float*)d_in[2];
  const float* t    = (const float*)d_in[3];
  const float* u    = (const float*)d_in[4];
  const float* v    = (const float*)d_in[5];
  const float* w    = (const float*)d_in[6];
  const float* Wf   = (const float*)d_in[7];
  const float* bf   = (const float*)d_in[8];
  const float* W1   = (const float*)d_in[9];
  const float* b1   = (const float*)d_in[10];
  const float* W2   = (const float*)d_in[11];
  const float* b2   = (const float*)d_in[12];
  const float* Wl   = (const float*)d_in[13];
  const float* bl   = (const float*)d_in[14];
  const float* lam1 = (const float*)d_in[15];
  const float* lam2 = (const float*)d_in[16];
  float* out = (float*)d_out;

  zero_loss_kernel<<<1, 32, 0, stream>>>(out + 4 * BATCH);
  pinn_jet_kernel<<<BATCH / SPG, 256, LDS_FLOATS * sizeof(float), stream>>>(
      x, y, z, t, u, v, w, Wf, bf, W1, b1, W2, b2, Wl, bl, lam1, lam2, out);
}